// Model_48533130445191
// MI455X (gfx1250) — compile-verified
//
#include <hip/hip_runtime.h>
#include <hip/hip_bf16.h>
#include <cstdint>
#include <cstddef>

#define FIN 256
#define FH2 128
#define FH  64
#define NC  10

typedef float v2f __attribute__((ext_vector_type(2)));
typedef float v8f __attribute__((ext_vector_type(8)));

// ---------------- degree / normalization ----------------

__global__ __launch_bounds__(256) void k_fill1(float* p, int n) {
    int i = blockIdx.x * blockDim.x + threadIdx.x;
    if (i < n) p[i] = 1.0f;   // self-loop contributes 1 to every degree
}

__global__ __launch_bounds__(256) void k_deg(const int* __restrict__ dst, float* deg, int nE) {
    int i = blockIdx.x * blockDim.x + threadIdx.x;
    if (i < nE) unsafeAtomicAdd(&deg[dst[i]], 1.0f);
}

__global__ __launch_bounds__(256) void k_rsqrt(float* p, int n) {
    int i = blockIdx.x * blockDim.x + threadIdx.x;
    if (i < n) p[i] = rsqrtf(p[i]);   // deg >= 1 always (self-loop)
}

// ---------------- W repack into WMMA-fragment-major layout ----------------
// rW[((k4*NT)+t)*64 + lane*2 + j] = W[(k4*4 + (lane>>4)*2 + j) * NDIM + t*16 + (lane&15)]
// so each lane's B-fragment (b.x = row ka, b.y = row ka+1) is one contiguous b64.

__global__ __launch_bounds__(256) void k_repack(const float* __restrict__ W,
                                                float* __restrict__ rW,
                                                int KDIM, int NDIM) {
    int idx = blockIdx.x * blockDim.x + threadIdx.x;
    if (idx >= KDIM * NDIM) return;
    int lanepair = idx & 63;          // lane*2 + j
    int lane = lanepair >> 1;
    int j    = lanepair & 1;
    int frag = idx >> 6;
    int nt   = NDIM >> 4;
    int t    = frag % nt;
    int k4   = frag / nt;
    int row  = k4 * 4 + (lane >> 4) * 2 + j;
    int col  = t * 16 + (lane & 15);
    rW[idx] = W[row * NDIM + col];
}

// ---------------- fp32 WMMA GEMM: Y[nrows x (NT*16)] = A[nrows x KDIM] @ W ----------------
// One wave computes a 16-row strip across all NT column tiles. Repacked W is staged
// into dynamic LDS; B-frags come from ds_load_2addr_b64 with explicit double-buffering
// so the DS latency overlaps a full k-step of WMMAs (no single-register recycling).
// A-frag (16x4 f32): lanes 0-15 hold K=k,k+1 ; lanes 16-31 hold K=k+2,k+3 (ISA 7.12.2).

template <int KDIM, int NT, bool RELU_A>
__global__ __launch_bounds__(256) void k_gemm_wmma(const float* __restrict__ A,
                                                   const float* __restrict__ rW,
                                                   float* __restrict__ Y, int nrows) {
    constexpr int NDIM  = NT * 16;
    constexpr int K4    = KDIM / 4;
    constexpr int FRAGS = K4 * NT;          // 64-float fragments
    static_assert((K4 & 1) == 0, "K4 must be even for 2-deep pipelining");
    extern __shared__ float sW[];           // FRAGS * 64 floats

    // cooperative stage of repacked W (float4-vectorized, coalesced)
    {
        const float4* s4 = (const float4*)rW;
        float4* d4 = (float4*)sW;
        for (int t = threadIdx.x; t < FRAGS * 16; t += blockDim.x) d4[t] = s4[t];
    }
    __syncthreads();

    int wave = (int)((blockIdx.x * blockDim.x + threadIdx.x) >> 5);
    int lane = threadIdx.x & 31;
    int m0 = wave * 16;
    if (m0 >= nrows) return;                // wave-uniform: EXEC stays all-ones for WMMA
    int half = lane >> 4;                   // 0: lanes 0-15, 1: lanes 16-31
    int lm   = lane & 15;

    const float* arow  = A + (size_t)(m0 + lm) * KDIM + half * 2;
    const float* sfrag = sW + lane * 2;

    v8f acc[NT] = {};
    v2f b0[NT], b1[NT];

    auto loadB = [&](v2f (&bf)[NT], int k4) {
        const float* p = sfrag + (size_t)k4 * NT * 64;
#pragma unroll
        for (int t = 0; t < NT; ++t) bf[t] = *(const v2f*)(p + t * 64);
    };
    auto loadA = [&](int k4) -> float2 {
        float2 av = *(const float2*)(arow + k4 * 4);   // one global_load_b64
        if (RELU_A) { av.x = fmaxf(av.x, 0.0f); av.y = fmaxf(av.y, 0.0f); }
        return av;
    };
    auto compute = [&](float2 av, v2f (&bf)[NT]) {
        v2f af; af.x = av.x; af.y = av.y;
#pragma unroll
        for (int t = 0; t < NT; ++t)
            acc[t] = __builtin_amdgcn_wmma_f32_16x16x4_f32(
                false, af, false, bf[t], (short)0, acc[t], false, false);
    };

    float2 a0 = loadA(0);
    loadB(b0, 0);
    float2 a1;
    for (int k4 = 0; k4 < K4; k4 += 2) {
        a1 = loadA(k4 + 1);
        loadB(b1, k4 + 1);
        compute(a0, b0);                     // overlaps with b1/a1 loads
        if (k4 + 2 < K4) {
            a0 = loadA(k4 + 2);
            loadB(b0, k4 + 2);
        }
        compute(a1, b1);                     // overlaps with b0/a0 loads
    }

    int mrow = m0 + half * 8;               // C/D: VGPR r -> row r (lanes 0-15) / row r+8
#pragma unroll
    for (int t = 0; t < NT; ++t) {
#pragma unroll
        for (int r = 0; r < 8; ++r) {
            Y[(size_t)(mrow + r) * NDIM + t * 16 + lm] = acc[t][r];
        }
    }
}

// ---------------- aggregation init: agg = bias + Y * dinv^2 (self loop) ----------------

__global__ __launch_bounds__(256) void k_init_agg(const float* __restrict__ Y,
                                                  const float* __restrict__ dinv,
                                                  const float* __restrict__ bias,
                                                  float* __restrict__ out, int n, int F) {
    long long idx = (long long)blockIdx.x * blockDim.x + threadIdx.x;
    if (idx >= (long long)n * F) return;
    int i = (int)(idx / F);
    int j = (int)(idx % F);
    float di = dinv[i];
    out[idx] = bias[j] + Y[idx] * di * di;
}

// ---------------- edge scatter: one wave per edge ----------------

__global__ __launch_bounds__(256) void k_scatter128(const int* __restrict__ src,
                                                    const int* __restrict__ dst,
                                                    const float* __restrict__ dinv,
                                                    const float* __restrict__ Y,
                                                    float* __restrict__ out, int nE) {
    int e = (int)((blockIdx.x * blockDim.x + threadIdx.x) >> 5);
    int lane = threadIdx.x & 31;
    if (e >= nE) return;
    int s = src[e], d = dst[e];
    float nrm = dinv[s] * dinv[d];
    float4 v = *(const float4*)(Y + (size_t)s * FH2 + lane * 4);
    float* o = out + (size_t)d * FH2 + lane * 4;
    unsafeAtomicAdd(o + 0, v.x * nrm);
    unsafeAtomicAdd(o + 1, v.y * nrm);
    unsafeAtomicAdd(o + 2, v.z * nrm);
    unsafeAtomicAdd(o + 3, v.w * nrm);
}

__global__ __launch_bounds__(256) void k_scatter64(const int* __restrict__ src,
                                                   const int* __restrict__ dst,
                                                   const float* __restrict__ dinv,
                                                   const float* __restrict__ Y,
                                                   float* __restrict__ out, int nE) {
    int e = (int)((blockIdx.x * blockDim.x + threadIdx.x) >> 5);
    int lane = threadIdx.x & 31;
    if (e >= nE) return;
    int s = src[e], d = dst[e];
    float nrm = dinv[s] * dinv[d];
    float2 v = *(const float2*)(Y + (size_t)s * FH + lane * 2);
    float* o = out + (size_t)d * FH + lane * 2;
    unsafeAtomicAdd(o + 0, v.x * nrm);
    unsafeAtomicAdd(o + 1, v.y * nrm);
}

// ---------------- head: logits = relu(h) @ Wl + bl, then log_softmax ----------------

__global__ __launch_bounds__(256) void k_head(const float* __restrict__ agg,
                                              const float* __restrict__ Wl,
                                              const float* __restrict__ bl,
                                              float* __restrict__ out, int n) {
    __shared__ float sWl[FH * NC];
    __shared__ float sB[NC];
    for (int t = threadIdx.x; t < FH * NC; t += blockDim.x) sWl[t] = Wl[t];
    if (threadIdx.x < NC) sB[threadIdx.x] = bl[threadIdx.x];
    __syncthreads();

    int i = blockIdx.x * blockDim.x + threadIdx.x;
    if (i >= n) return;

    float logit[NC];
#pragma unroll
    for (int c = 0; c < NC; ++c) logit[c] = sB[c];

    const float* row = agg + (size_t)i * FH;
    for (int j = 0; j < FH; ++j) {
        float hj = fmaxf(row[j], 0.0f);
#pragma unroll
        for (int c = 0; c < NC; ++c) logit[c] += hj * sWl[j * NC + c];
    }

    float mx = logit[0];
#pragma unroll
    for (int c = 1; c < NC; ++c) mx = fmaxf(mx, logit[c]);
    float sum = 0.0f;
#pragma unroll
    for (int c = 0; c < NC; ++c) sum += __expf(logit[c] - mx);
    float lse = mx + __logf(sum);
#pragma unroll
    for (int c = 0; c < NC; ++c) out[(size_t)i * NC + c] = logit[c] - lse;
}

// ---------------- host-side launch ----------------

extern "C" void kernel_launch(void* const* d_in, const int* in_sizes, int n_in,
                              void* d_out, int out_size, void* d_ws, size_t ws_size,
                              hipStream_t stream) {
    (void)n_in; (void)out_size; (void)ws_size;
    const float* x  = (const float*)d_in[0];
    const int*   ei = (const int*)d_in[1];
    const float* W1 = (const float*)d_in[2];
    const float* b1 = (const float*)d_in[3];
    const float* W2 = (const float*)d_in[4];
    const float* b2 = (const float*)d_in[5];
    const float* Wl = (const float*)d_in[6];
    const float* bl = (const float*)d_in[7];
    float* out = (float*)d_out;

    const int N = in_sizes[0] / FIN;
    const int E = in_sizes[1] / 2;
    const int* src = ei;        // edge_index[0]
    const int* dst = ei + E;    // edge_index[1]

    float* dinv = (float*)d_ws;               // N
    float* bufA = dinv + N;                   // N*128 (GEMM outputs)
    float* bufB = bufA + (size_t)N * FH2;     // N*128 (aggregation targets)
    float* rW1  = bufB + (size_t)N * FH2;     // FIN*FH2 repacked
    float* rW2  = rW1 + FIN * FH2;            // FH2*FH repacked

    const dim3 blk(256);
    const int nwaves = (N + 15) / 16;
    const unsigned gemm_blocks = (unsigned)(((long long)nwaves * 32 + 255) / 256);
    const unsigned edge_blocks = (unsigned)(((long long)E * 32 + 255) / 256);
    const size_t lds1 = (size_t)(FIN / 4) * (FH2 / 16) * 64 * sizeof(float);  // 128 KB
    const size_t lds2 = (size_t)(FH2 / 4) * (FH / 16) * 64 * sizeof(float);   //  32 KB

    // normalization
    k_fill1 <<<(N + 255) / 256, blk, 0, stream>>>(dinv, N);
    k_deg   <<<(E + 255) / 256, blk, 0, stream>>>(dst, dinv, E);
    k_rsqrt <<<(N + 255) / 256, blk, 0, stream>>>(dinv, N);

    // repack weights into fragment-major layout
    k_repack<<<(FIN * FH2 + 255) / 256, blk, 0, stream>>>(W1, rW1, FIN, FH2);
    k_repack<<<(FH2 * FH + 255) / 256, blk, 0, stream>>>(W2, rW2, FH2, FH);

    // layer 1: Y1 = x @ W1 ; agg1 = b1 + D^-1/2 A D^-1/2 Y1
    k_gemm_wmma<FIN, FH2 / 16, false><<<gemm_blocks, blk, lds1, stream>>>(x, rW1, bufA, N);
    k_init_agg<<<(unsigned)(((long long)N * FH2 + 255) / 256), blk, 0, stream>>>(bufA, dinv, b1, bufB, N, FH2);
    k_scatter128<<<edge_blocks, blk, 0, stream>>>(src, dst, dinv, bufA, bufB, E);

    // layer 2: Y2 = relu(agg1) @ W2 ; agg2 = b2 + D^-1/2 A D^-1/2 Y2
    k_gemm_wmma<FH2, FH / 16, true><<<gemm_blocks, blk, lds2, stream>>>(bufB, rW2, bufA, N);
    k_init_agg<<<(unsigned)(((long long)N * FH + 255) / 256), blk, 0, stream>>>(bufA, dinv, b2, bufB, N, FH);
    k_scatter64<<<edge_blocks, blk, 0, stream>>>(src, dst, dinv, bufA, bufB, E);

    // head + log_softmax
    k_head<<<(N + 255) / 256, blk, 0, stream>>>(bufB, Wl, bl, out, N);
}